// VanillaRNN_11493332484726
// MI455X (gfx1250) — compile-verified
//
#include <hip/hip_runtime.h>
#include <hip/hip_bf16.h>
#include <math.h>

// Problem constants (match reference)
constexpr int BATCH = 64;
constexpr int SEQ   = 512;
constexpr int HID   = 1024;
constexpr int EMB   = 1024;

typedef __attribute__((ext_vector_type(2))) float v2f;
typedef __attribute__((ext_vector_type(4))) float v4f;
typedef __attribute__((ext_vector_type(8))) float v8f;
typedef __attribute__((ext_vector_type(4))) int   v4i;

// Async global->LDS path (ASYNCcnt) if this toolchain declares the builtins.
#if __has_builtin(__builtin_amdgcn_global_load_async_to_lds_b128) && \
    __has_builtin(__builtin_amdgcn_s_wait_asynccnt)
#define VRNN_HAS_ASYNC_LDS 1
typedef __attribute__((address_space(1))) v4i* g1_v4i_ptr;   // global int4*
typedef __attribute__((address_space(3))) v4i* l3_v4i_ptr;   // LDS int4*
#else
#define VRNN_HAS_ASYNC_LDS 0
#endif

// =========================================================================
// Kernel 1: xproj[m, n] = sum_e emb_eff[tok(m), e] * Wx[n, e] + bx[n]
//   m = s*BATCH + b  (row-major [S,B,H] output), tok(m) = seq[b*SEQ + s]
//   PAD row (token 0) contributes zero embedding.
// Block: 128 threads = 4 wave32. Tile: 16 (M) x 64 (N). K loop in 64-chunks.
// Each wave computes one 16x16 tile via V_WMMA_F32_16X16X4_F32.
// =========================================================================
constexpr int KCH1 = 64;

__global__ __launch_bounds__(128)
void vrnn_xproj_kernel(const int* __restrict__ seq,
                       const float* __restrict__ emb,
                       const float* __restrict__ Wx,
                       const float* __restrict__ bx,
                       float* __restrict__ xproj)
{
    __shared__ float As[16][KCH1 + 1];   // gathered embedding rows
    __shared__ float Bs[64][KCH1 + 1];   // Wx rows (B matrix = Wx^T)
    __shared__ int   toks[16];

    const int tid   = threadIdx.x;
    const int lane  = tid & 31;
    const int wave  = tid >> 5;
    const int mBase = blockIdx.x * 16;       // row tile of M = S*B
    const int nBase = blockIdx.y * 64;       // col tile of H
    const int nWave = nBase + wave * 16;

    if (tid < 16) {
        int m = mBase + tid;
        int s = m / BATCH;
        int b = m % BATCH;
        toks[tid] = seq[b * SEQ + s];
    }
    __syncthreads();

    v8f acc;
    {
        float bias = bx[nWave + (lane & 15)];
        #pragma unroll
        for (int r = 0; r < 8; ++r) acc[r] = bias;
    }

    const int mr = lane & 15;
    const int kk = (lane >> 4) * 2;

    for (int k0 = 0; k0 < EMB; k0 += KCH1) {
        for (int i = tid; i < 16 * KCH1; i += 128) {
            int m = i / KCH1, e = i % KCH1;
            int tok = toks[m];
            As[m][e] = (tok == 0) ? 0.0f
                                  : emb[(size_t)tok * EMB + k0 + e];
        }
        for (int i = tid; i < 64 * KCH1; i += 128) {
            int n = i / KCH1, e = i % KCH1;
            Bs[n][e] = Wx[(size_t)(nBase + n) * EMB + k0 + e];
        }
        __syncthreads();

        #pragma unroll
        for (int k = 0; k < KCH1; k += 4) {
            v2f a, b;
            a.x = As[mr][k + kk];
            a.y = As[mr][k + kk + 1];
            b.x = Bs[wave * 16 + mr][k + kk];
            b.y = Bs[wave * 16 + mr][k + kk + 1];
            acc = __builtin_amdgcn_wmma_f32_16x16x4_f32(
                      false, a, false, b, (short)0, acc, false, false);
        }
        __syncthreads();
    }

    {
        const int n   = nWave + (lane & 15);
        const int mHi = (lane >> 4) * 8;
        #pragma unroll
        for (int r = 0; r < 8; ++r) {
            int m = mBase + r + mHi;
            xproj[(size_t)m * HID + n] = acc[r];
        }
    }
}

// =========================================================================
// Kernel 2: persistent scan. 16 blocks (one per 64-col slice of H),
// 512 threads = 16 wave32. Wh slice (64x1024 fp32) lives in LDS for the
// whole kernel; per step the previous h (L2-resident) is staged with
// DOUBLE-BUFFERED async global->LDS B128 transfers (ASYNCcnt), overlapping
// chunk c+1's DMA with chunk c's WMMA chain. Device-wide sync via a
// monotonic atomic-counter barrier (zeroed every call).
//
// LDS bank math (64 banks): strides 1028 and 68 are == 4 (mod 64) and
// 16B-aligned -> fragment reads {4n+k,4n+k+1}/{4n+k+2,4n+k+3} are
// conflict-free and b128 transfers are aligned.
// Total LDS: (64*1028 + 2*64*68)*4 = 297,984 B < 320 KB per WGP.
// =========================================================================
constexpr int SCAN_BLOCKS  = HID / 64;   // 16
constexpr int SCAN_THREADS = 512;        // 16 waves
constexpr int WH_STRIDE    = 1028;       // HID + 4
constexpr int KCH2         = 64;
constexpr int AS_STRIDE    = 68;         // KCH2 + 4
constexpr int NCHUNK       = HID / KCH2; // 16
constexpr size_t SCAN_LDS_BYTES =
    (size_t)(64 * WH_STRIDE + 2 * 64 * AS_STRIDE) * sizeof(float);

// Stage one 64 x KCH2 chunk of h_prev (row stride HID) into an LDS buffer.
// 1024 float4 transfers, 2 per thread, coalesced.
__device__ __forceinline__
void vrnn_stage_chunk(const float* __restrict__ gsrc,   // hprev + k0
                      float* __restrict__ lbuf, int tid)
{
    #pragma unroll
    for (int i = tid; i < 64 * (KCH2 / 4); i += SCAN_THREADS) {
        int row = i >> 4;                 // / 16
        int c4  = i & 15;
        const float* g = gsrc + (size_t)row * HID + c4 * 4;
        float*       l = lbuf + row * AS_STRIDE + c4 * 4;
#if VRNN_HAS_ASYNC_LDS
        __builtin_amdgcn_global_load_async_to_lds_b128(
            (g1_v4i_ptr)(v4i*)g,          // global src (AS1 int4*, const cast away)
            (l3_v4i_ptr)(v4i*)l,          // LDS dst (AS3 int4*)
            /*offset=*/0, /*cpol=*/0);
#else
        *(v4f*)l = *(const v4f*)g;
#endif
    }
}

__device__ __forceinline__ void vrnn_wait_stage()
{
#if VRNN_HAS_ASYNC_LDS
    __builtin_amdgcn_s_wait_asynccnt(0);
#endif
}

__global__ __launch_bounds__(SCAN_THREADS)
void vrnn_scan_kernel(const float* __restrict__ Wh,
                      const float* __restrict__ bh,
                      const float* __restrict__ xproj,
                      float* __restrict__ out,
                      int* __restrict__ barrier_ctr)
{
    extern __shared__ float lds[];
    float* Whs  = lds;                           // [64][WH_STRIDE]
    float* Ahs0 = lds + 64 * WH_STRIDE;          // [64][AS_STRIDE]
    float* Ahs1 = Ahs0 + 64 * AS_STRIDE;         // [64][AS_STRIDE]

    const int tid   = threadIdx.x;
    const int lane  = tid & 31;
    const int wave  = tid >> 5;
    const int mTile = (wave & 3) * 16;           // batch-row tile (local)
    const int nTile = (wave >> 2) * 16;          // col tile within slice
    const int nBase = blockIdx.x * 64;
    const int mr    = lane & 15;
    const int kk    = (lane >> 4) * 2;
    const int mHi   = (lane >> 4) * 8;

    // ---- one-time: stage Wh slice [nBase .. nBase+63][:] into LDS ----
    for (int i = tid; i < 64 * (HID / 4); i += SCAN_THREADS) {
        int row = i >> 8;                        // / 256
        int c4  = i & 255;
        v4f v = *(const v4f*)&Wh[(size_t)(nBase + row) * HID + c4 * 4];
        *(v4f*)&Whs[row * WH_STRIDE + c4 * 4] = v;
    }

    const int   nGlob = nBase + nTile + mr;
    const float bias  = bh[nGlob];
    const float* Brow = &Whs[(nTile + mr) * WH_STRIDE];

    __syncthreads();                             // Whs ready

    for (int s = 0; s < SEQ; ++s) {
        v8f acc = {};
        if (s > 0) {
            const float* hprev = out + (size_t)(s - 1) * BATCH * HID;

            vrnn_stage_chunk(hprev, Ahs0, tid);          // chunk 0 -> buf0
            for (int c = 0; c < NCHUNK; ++c) {
                float* cur = (c & 1) ? Ahs1 : Ahs0;
                float* nxt = (c & 1) ? Ahs0 : Ahs1;

                vrnn_wait_stage();      // own chunk-c loads complete
                __syncthreads();        // all waves: chunk c in LDS AND
                                        // chunk c-1 compute finished
                if (c + 1 < NCHUNK)     // overlap: DMA chunk c+1 during compute
                    vrnn_stage_chunk(hprev + (c + 1) * KCH2, nxt, tid);

                const float* Ar = cur + (mTile + mr) * AS_STRIDE;
                const float* Bk = Brow + c * KCH2;
                #pragma unroll
                for (int k = 0; k < KCH2; k += 4) {
                    v2f a, b;
                    a.x = Ar[k + kk];
                    a.y = Ar[k + kk + 1];
                    b.x = Bk[k + kk];
                    b.y = Bk[k + kk + 1];
                    acc = __builtin_amdgcn_wmma_f32_16x16x4_f32(
                              false, a, false, b, (short)0, acc, false, false);
                }
            }
        }

        // ---- epilogue: tanh(acc + bh + xproj[s]) -> out[s] ----
        const float* xp = xproj + (size_t)s * BATCH * HID;
        float*       ho = out   + (size_t)s * BATCH * HID;
        #pragma unroll
        for (int r = 0; r < 8; ++r) {
            int m = mTile + mHi + r;
            float v = acc[r] + bias + xp[(size_t)m * HID + nGlob];
            ho[(size_t)m * HID + nGlob] = tanhf(v);
        }

        // ---- device-wide barrier (monotonic counter; zeroed per call) ----
        __threadfence();                         // h writes agent-visible
        __syncthreads();
        if (tid == 0) {
            __hip_atomic_fetch_add(barrier_ctr, 1, __ATOMIC_ACQ_REL,
                                   __HIP_MEMORY_SCOPE_AGENT);
            const int target = SCAN_BLOCKS * (s + 1);
            while (__hip_atomic_load(barrier_ctr, __ATOMIC_ACQUIRE,
                                     __HIP_MEMORY_SCOPE_AGENT) < target) {
                __builtin_amdgcn_s_sleep(1);
            }
        }
        __syncthreads();
    }
}

// =========================================================================
// Launch: counter memset (capture-safe) -> xproj GEMM -> persistent scan.
// Workspace layout: [0,256): barrier counter; [256, 256+134MB): xproj.
// =========================================================================
extern "C" void kernel_launch(void* const* d_in, const int* in_sizes, int n_in,
                              void* d_out, int out_size, void* d_ws, size_t ws_size,
                              hipStream_t stream)
{
    (void)in_sizes; (void)n_in; (void)out_size; (void)ws_size;

    const int*   seq = (const int*)  d_in[0];   // [B,S]
    const float* emb = (const float*)d_in[1];   // [V,E]
    const float* Wh  = (const float*)d_in[2];   // [H,H]
    const float* bh  = (const float*)d_in[3];   // [H]
    const float* Wx  = (const float*)d_in[4];   // [H,E]
    const float* bx  = (const float*)d_in[5];   // [H]

    float* out   = (float*)d_out;                       // [S,B,H]
    int*   ctr   = (int*)d_ws;                          // barrier counter
    float* xproj = (float*)((char*)d_ws + 256);         // [S,B,H] scratch

    // zero the barrier counter every call (deterministic across replays)
    (void)hipMemsetAsync(d_ws, 0, 256, stream);

    // allow > 64KB dynamic LDS for the persistent scan kernel
    (void)hipFuncSetAttribute((const void*)vrnn_scan_kernel,
                              hipFuncAttributeMaxDynamicSharedMemorySize,
                              (int)SCAN_LDS_BYTES);

    // Phase 1: xproj = gather(emb) . Wx^T + bx   (M=32768, N=1024, K=1024)
    {
        dim3 grid((SEQ * BATCH) / 16, HID / 64);
        vrnn_xproj_kernel<<<grid, 128, 0, stream>>>(seq, emb, Wx, bx, xproj);
    }

    // Phase 2: persistent scan over all 512 steps (single launch)
    vrnn_scan_kernel<<<dim3(SCAN_BLOCKS), SCAN_THREADS, SCAN_LDS_BYTES, stream>>>(
        Wh, bh, xproj, out, ctr);
}